// RegimeSwitchingKF_7464653161148
// MI455X (gfx1250) — compile-verified
//
#include <hip/hip_runtime.h>

#define NSUBJ   8192
#define TSTEPS  512
#define OBS     9
#define BLOCK   256
#define NBLOCKS (NSUBJ / BLOCK)
#define LOG2PI_F 1.8378770664093453f

typedef unsigned int u32;
typedef unsigned long long u64;
typedef __attribute__((ext_vector_type(4))) u32 u32x4;
typedef __attribute__((ext_vector_type(8))) u32 u32x8;
typedef __attribute__((ext_vector_type(2))) float v2f;
typedef __attribute__((ext_vector_type(8))) float v8f;

// ---------------------------------------------------------------------------
// Tensor Data Mover: stage `rows` rows of `rowlen` f32 elements (row stride
// `stride` f32 elements) from global memory into LDS at byte offset lds_off.
// D# built per CDNA5 ISA ch.8: group0 (4 SGPRs) + group1 (8 SGPRs), 2D tile,
// groups 2/3 omitted (NULL). Tracked by TENSORcnt.
// ---------------------------------------------------------------------------
__device__ __forceinline__ void tdm_load_rows_f32(u32 lds_off, u64 gaddr,
                                                  u32 rows, u32 rowlen,
                                                  u32 stride) {
  u32x4 g0;
  g0[0] = 1u;                                    // count=1 (valid user D#)
  g0[1] = lds_off;                               // LDS byte address
  g0[2] = (u32)gaddr;                            // global_addr[31:0]
  g0[3] = ((u32)(gaddr >> 32) & 0x01FFFFFFu)     // global_addr[56:32]
          | (2u << 30);                          // type=2 ("image")
  u32x8 g1;
  g1[0] = 2u << 16;                              // data_size=2 -> 4 bytes
  g1[1] = (rowlen & 0xFFFFu) << 16;              // tensor_dim0[15:0]
  g1[2] = (rowlen >> 16) | ((rows & 0xFFFFu) << 16); // tdim0 hi | tdim1 lo
  g1[3] = (rows >> 16) | ((rowlen & 0xFFFFu) << 16); // tdim1 hi | tile_dim0
  g1[4] = rows & 0xFFFFu;                        // tile_dim1 | tile_dim2=0
  g1[5] = stride;                                // tensor_dim0_stride[31:0]
  g1[6] = 0u;                                    // stride hi | dim1_stride lo
  g1[7] = 0u;
  asm volatile("tensor_load_to_lds %0, %1" :: "s"(g0), "s"(g1) : "memory");
}

__device__ __forceinline__ void wait_tensorcnt_le1() {
  asm volatile("s_wait_tensorcnt 0x1" ::: "memory");
}
__device__ __forceinline__ void wait_tensorcnt_0() {
  asm volatile("s_wait_tensorcnt 0x0" ::: "memory");
}

// Symmetric 3x3 inverse via cofactors; returns determinant.
__device__ __forceinline__ float sym3_inv(float a00, float a01, float a02,
                                          float a11, float a12, float a22,
                                          float& i00, float& i01, float& i02,
                                          float& i11, float& i12, float& i22) {
  float c00 = a11 * a22 - a12 * a12;
  float c01 = a02 * a12 - a01 * a22;
  float c02 = a01 * a12 - a02 * a11;
  float det = a00 * c00 + a01 * c01 + a02 * c02;
  float id  = 1.0f / det;
  i00 = c00 * id; i01 = c01 * id; i02 = c02 * id;
  i11 = (a00 * a22 - a02 * a02) * id;
  i12 = (a01 * a02 - a00 * a12) * id;
  i22 = (a00 * a11 - a01 * a01) * id;
  return det;
}

__global__ __launch_bounds__(BLOCK) void rskf_main(
    const float* __restrict__ y,
    const float* __restrict__ b1_r1, const float* __restrict__ lam1f,
    const float* __restrict__ q1d,   const float* __restrict__ B2in,
    const float* __restrict__ lam2f, const float* __restrict__ q2d,
    const float* __restrict__ rd,    const float* __restrict__ g1p,
    const float* __restrict__ g2p,   const float* __restrict__ g3p,
    const float* __restrict__ g4p,   float* __restrict__ partials) {
  __shared__ float ybuf[2][BLOCK][OBS];   // double-buffered TDM staging
  __shared__ float red[BLOCK];

  const int tid = threadIdx.x;
  const int wid = tid >> 5;               // wave index within block (wave32)

  // ---- load shared parameters (uniform; scalar-cached) ----
  const float b0 = b1_r1[0], b1 = b1_r1[1], b2 = b1_r1[2];
  const float l0 = lam1f[0], l1 = lam1f[1], l2 = lam1f[2];
  const float l3 = lam1f[3], l4 = lam1f[4], l5 = lam1f[5];
  const float q10 = fabsf(q1d[0]) + 1e-4f, q11 = fabsf(q1d[1]) + 1e-4f,
              q12 = fabsf(q1d[2]) + 1e-4f;
  const float B200 = B2in[0], B201 = B2in[1], B202 = B2in[2];
  const float B210 = B2in[3], B211 = B2in[4], B212 = B2in[5];
  const float B220 = B2in[6], B221 = B2in[7], B222 = B2in[8];
  const float m1c = lam2f[0], m2c = lam2f[1], m3c = lam2f[2], m4c = lam2f[3],
              m5c = lam2f[4], m6c = lam2f[5], m7c = lam2f[6], m8c = lam2f[7];
  const float q20 = fabsf(q2d[0]) + 1e-4f, q21 = fabsf(q2d[1]) + 1e-4f,
              q22 = fabsf(q2d[2]) + 1e-4f;
  float rr[OBS], ri[OBS];
  float logdetR = 0.0f;
#pragma unroll
  for (int i = 0; i < OBS; ++i) {
    rr[i] = fabsf(rd[i]) + 1e-4f;
    ri[i] = 1.0f / rr[i];
    logdetR += __logf(rr[i]);
  }
  // G1 = Lam1^T R^-1 Lam1 (diagonal, by 3-factor block structure)
  const float G10 = ri[0] + ri[1] * l0 * l0 + ri[2] * l1 * l1;
  const float G11 = ri[3] + ri[4] * l2 * l2 + ri[5] * l3 * l3;
  const float G12 = ri[6] + ri[7] * l4 * l4 + ri[8] * l5 * l5;
  // g2s = (Lam2^T R^-1 Lam2)[0,0] (only nonzero entry; rank-1 loading)
  const float g2s = ri[0] + ri[1] * m1c * m1c + ri[2] * m2c * m2c +
                    ri[3] * m3c * m3c + ri[4] * m4c * m4c + ri[5] * m5c * m5c +
                    ri[6] * m6c * m6c + ri[7] * m7c * m7c + ri[8] * m8c * m8c;
  const float ga1 = g1p[0], ga20 = g2p[0], ga21 = g2p[1], ga22 = g2p[2];
  const float ga3 = g3p[0], ga40 = g4p[0], ga41 = g4p[1], ga42 = g4p[2];

  // ---- per-subject state (registers) ----
  float e10 = 0.f, e11 = 0.f, e12 = 0.f;
  float P100 = 1000.f, P101 = 0.f, P102 = 0.f, P111 = 1000.f, P112 = 0.f,
        P122 = 1000.f;
  float e20 = 0.f, e21 = 0.f, e22 = 0.f;
  float P200 = 1000.f, P201 = 0.f, P202 = 0.f, P211 = 1000.f, P212 = 0.f,
        P222 = 1000.f;
  float mr0 = 0.99f, mr1 = 0.01f, llacc = 0.0f;

  // ---- TDM staging setup: each wave DMAs its own 32 subjects' y rows ----
  const u32 lds0      = (u32)(size_t)(&ybuf[0][0][0]);
  const u32 sliceOff  = (u32)(wid * 32 * OBS * 4);
  const u32 slotBytes = (u32)(BLOCK * OBS * 4);
  const u64 gbase = (u64)(size_t)y +
                    (u64)(blockIdx.x * BLOCK + wid * 32) * (TSTEPS * OBS * 4);
  tdm_load_rows_f32(lds0 + sliceOff, gbase, 32, OBS, TSTEPS * OBS);

  for (int t = 0; t < TSTEPS; ++t) {
    if (t + 1 < TSTEPS) {   // prefetch step t+1 while step t computes
      tdm_load_rows_f32(lds0 + (u32)(((t + 1) & 1)) * slotBytes + sliceOff,
                        gbase + (u64)(t + 1) * (OBS * 4), 32, OBS,
                        TSTEPS * OBS);
      wait_tensorcnt_le1();   // tensor ops complete in order -> slot t ready
    } else {
      wait_tensorcnt_0();
    }
    const float* yt = &ybuf[t & 1][tid][0];
    const float y0 = yt[0], y1 = yt[1], y2 = yt[2], y3 = yt[3], y4 = yt[4],
                y5 = yt[5], y6 = yt[6], y7 = yt[7], y8 = yt[8];

    // ---- regime transition probs from previous filtered means ----
    const float x1  = ga1 + ga20 * e10 + ga21 * e11 + ga22 * e12;
    const float x2  = ga3 + ga40 * e20 + ga41 * e21 + ga42 * e22;
    const float p11 = 1.0f / (1.0f + __expf(-x1));
    const float p22 = 1.0f / (1.0f + __expf(-x2));
    const float mp0 = p11 * mr0 + (1.0f - p22) * mr1;
    const float mp1 = (1.0f - p11) * mr0 + p22 * mr1;

    // ================= regime 1 (diag B, 3-factor CFA) =================
    float ll1;
    {
      const float ep0 = b0 * e10, ep1 = b1 * e11, ep2 = b2 * e12;
      const float Pp00 = b0 * b0 * P100 + q10;
      const float Pp01 = b0 * b1 * P101;
      const float Pp02 = b0 * b2 * P102;
      const float Pp11 = b1 * b1 * P111 + q11;
      const float Pp12 = b1 * b2 * P112;
      const float Pp22 = b2 * b2 * P122 + q12;
      const float v0 = y0 - ep0,      v1 = y1 - l0 * ep0, v2 = y2 - l1 * ep0;
      const float v3 = y3 - ep1,      v4 = y4 - l2 * ep1, v5 = y5 - l3 * ep1;
      const float v6 = y6 - ep2,      v7 = y7 - l4 * ep2, v8 = y8 - l5 * ep2;
      const float w0 = v0 * ri[0], w1 = v1 * ri[1], w2 = v2 * ri[2];
      const float w3 = v3 * ri[3], w4 = v4 * ri[4], w5 = v5 * ri[5];
      const float w6 = v6 * ri[6], w7 = v7 * ri[7], w8 = v8 * ri[8];
      const float quadw = v0 * w0 + v1 * w1 + v2 * w2 + v3 * w3 + v4 * w4 +
                          v5 * w5 + v6 * w6 + v7 * w7 + v8 * w8;
      const float t30 = w0 + l0 * w1 + l1 * w2;
      const float t31 = w3 + l2 * w4 + l3 * w5;
      const float t32 = w6 + l4 * w7 + l5 * w8;
      float iP00, iP01, iP02, iP11, iP12, iP22;
      const float detP = sym3_inv(Pp00, Pp01, Pp02, Pp11, Pp12, Pp22,
                                  iP00, iP01, iP02, iP11, iP12, iP22);
      float Pf00, Pf01, Pf02, Pf11, Pf12, Pf22;   // P_f = S^-1
      const float detS = sym3_inv(iP00 + G10, iP01, iP02, iP11 + G11, iP12,
                                  iP22 + G12, Pf00, Pf01, Pf02, Pf11, Pf12,
                                  Pf22);
      const float kv0 = Pf00 * t30 + Pf01 * t31 + Pf02 * t32;
      const float kv1 = Pf01 * t30 + Pf11 * t31 + Pf12 * t32;
      const float kv2 = Pf02 * t30 + Pf12 * t31 + Pf22 * t32;
      const float quad = quadw - (t30 * kv0 + t31 * kv1 + t32 * kv2);
      ll1 = -0.5f * ((float)OBS * LOG2PI_F + logdetR + __logf(detP) +
                     __logf(detS) + quad);
      e10 = ep0 + kv0; e11 = ep1 + kv1; e12 = ep2 + kv2;
      P100 = Pf00; P101 = Pf01; P102 = Pf02; P111 = Pf11; P112 = Pf12;
      P122 = Pf22;
    }

    // ============== regime 2 (full B, rank-1 loading column) ==============
    float ll2;
    {
      const float ep0 = B200 * e20 + B201 * e21 + B202 * e22;
      const float ep1 = B210 * e20 + B211 * e21 + B212 * e22;
      const float ep2 = B220 * e20 + B221 * e21 + B222 * e22;
      const float M00 = B200 * P200 + B201 * P201 + B202 * P202;
      const float M01 = B200 * P201 + B201 * P211 + B202 * P212;
      const float M02 = B200 * P202 + B201 * P212 + B202 * P222;
      const float M10 = B210 * P200 + B211 * P201 + B212 * P202;
      const float M11 = B210 * P201 + B211 * P211 + B212 * P212;
      const float M12 = B210 * P202 + B211 * P212 + B212 * P222;
      const float M20 = B220 * P200 + B221 * P201 + B222 * P202;
      const float M21 = B220 * P201 + B221 * P211 + B222 * P212;
      const float M22 = B220 * P202 + B221 * P212 + B222 * P222;
      const float Pp00 = M00 * B200 + M01 * B201 + M02 * B202 + q20;
      const float Pp01 = M00 * B210 + M01 * B211 + M02 * B212;
      const float Pp02 = M00 * B220 + M01 * B221 + M02 * B222;
      const float Pp11 = M10 * B210 + M11 * B211 + M12 * B212 + q21;
      const float Pp12 = M10 * B220 + M11 * B221 + M12 * B222;
      const float Pp22 = M20 * B220 + M21 * B221 + M22 * B222 + q22;
      const float v0 = y0 - ep0;
      const float v1 = y1 - m1c * ep0, v2 = y2 - m2c * ep0;
      const float v3 = y3 - m3c * ep0, v4 = y4 - m4c * ep0;
      const float v5 = y5 - m5c * ep0, v6 = y6 - m6c * ep0;
      const float v7 = y7 - m7c * ep0, v8 = y8 - m8c * ep0;
      const float w0 = v0 * ri[0], w1 = v1 * ri[1], w2 = v2 * ri[2];
      const float w3 = v3 * ri[3], w4 = v4 * ri[4], w5 = v5 * ri[5];
      const float w6 = v6 * ri[6], w7 = v7 * ri[7], w8 = v8 * ri[8];
      const float quadw = v0 * w0 + v1 * w1 + v2 * w2 + v3 * w3 + v4 * w4 +
                          v5 * w5 + v6 * w6 + v7 * w7 + v8 * w8;
      const float t30 = w0 + m1c * w1 + m2c * w2 + m3c * w3 + m4c * w4 +
                        m5c * w5 + m6c * w6 + m7c * w7 + m8c * w8;
      float iP00, iP01, iP02, iP11, iP12, iP22;
      const float detP = sym3_inv(Pp00, Pp01, Pp02, Pp11, Pp12, Pp22,
                                  iP00, iP01, iP02, iP11, iP12, iP22);
      float Pf00, Pf01, Pf02, Pf11, Pf12, Pf22;   // P_f = S^-1
      const float detS = sym3_inv(iP00 + g2s, iP01, iP02, iP11, iP12, iP22,
                                  Pf00, Pf01, Pf02, Pf11, Pf12, Pf22);
      const float kv0 = Pf00 * t30, kv1 = Pf01 * t30, kv2 = Pf02 * t30;
      const float quad = quadw - t30 * kv0;
      ll2 = -0.5f * ((float)OBS * LOG2PI_F + logdetR + __logf(detP) +
                     __logf(detS) + quad);
      e20 = ep0 + kv0; e21 = ep1 + kv1; e22 = ep2 + kv2;
      P200 = Pf00; P201 = Pf01; P202 = Pf02; P211 = Pf11; P212 = Pf12;
      P222 = Pf22;
    }

    // ---- mixing: log-sum-exp over regimes, update regime probs ----
    const float lj0 = ll1 + __logf(mp0 + 1e-9f);
    const float lj1 = ll2 + __logf(mp1 + 1e-9f);
    const float mx  = fmaxf(lj0, lj1);
    const float lm  = mx + __logf(__expf(lj0 - mx) + __expf(lj1 - mx));
    mr0 = __expf(lj0 - lm);
    mr1 = __expf(lj1 - lm);
    llacc += lm;
  }

  // ---- wave-level reduction via matrix core: D = ones(16x4) x B ----
  // Every lane deposits llacc in B's VGPR0 and 0 in VGPR1; A is all-ones in
  // both VGPRs (all-ones under any striping). Column j of D then sums the
  // values of lanes {j, j+16} regardless of which K-rows VGPR0 carries.
  // EXEC is all-1s here (uniform control flow), as WMMA requires.
  {
    v2f Am; Am[0] = 1.0f; Am[1] = 1.0f;
    v2f Bm; Bm[0] = llacc; Bm[1] = 0.0f;
    v8f Cz = {};
    v8f D = __builtin_amdgcn_wmma_f32_16x16x4_f32(
        /*neg_a=*/false, Am, /*neg_b=*/false, Bm,
        /*c_mod=*/(short)0, Cz, /*reuse_a=*/false, /*reuse_b=*/false);
    // D[0] on lane j (and j+16) = llacc[j] + llacc[j+16], j = lane & 15.
    red[tid] = ((tid & 31) < 16) ? D[0] : 0.0f;
  }
  __syncthreads();
#pragma unroll
  for (int s = BLOCK / 2; s > 0; s >>= 1) {
    if (tid < s) red[tid] += red[tid + s];
    __syncthreads();
  }
  if (tid == 0) partials[blockIdx.x] = red[0];
}

__global__ void rskf_reduce(const float* __restrict__ partials,
                            float* __restrict__ out) {
  if (threadIdx.x == 0 && blockIdx.x == 0) {
    float s = 0.0f;
    for (int i = 0; i < NBLOCKS; ++i) s += partials[i];  // fixed order
    out[0] = -s;   // negative log-likelihood
  }
}

extern "C" void kernel_launch(void* const* d_in, const int* in_sizes, int n_in,
                              void* d_out, int out_size, void* d_ws,
                              size_t ws_size, hipStream_t stream) {
  (void)in_sizes; (void)n_in; (void)out_size; (void)ws_size;
  const float* y      = (const float*)d_in[0];
  const float* b1r1   = (const float*)d_in[1];
  const float* lam1f  = (const float*)d_in[2];
  const float* q1d    = (const float*)d_in[3];
  const float* b1r2   = (const float*)d_in[4];
  const float* lam2f  = (const float*)d_in[5];
  const float* q2d    = (const float*)d_in[6];
  const float* rdiag  = (const float*)d_in[7];
  const float* gamma1 = (const float*)d_in[8];
  const float* gamma2 = (const float*)d_in[9];
  const float* gamma3 = (const float*)d_in[10];
  const float* gamma4 = (const float*)d_in[11];
  float* partials = (float*)d_ws;

  rskf_main<<<NBLOCKS, BLOCK, 0, stream>>>(y, b1r1, lam1f, q1d, b1r2, lam2f,
                                           q2d, rdiag, gamma1, gamma2, gamma3,
                                           gamma4, partials);
  rskf_reduce<<<1, 32, 0, stream>>>(partials, (float*)d_out);
}